// MyOutGCN_30167850287800
// MI455X (gfx1250) — compile-verified
//
#include <hip/hip_runtime.h>
#include <math.h>

typedef __attribute__((ext_vector_type(2))) float v2f;
typedef __attribute__((ext_vector_type(8))) float v8f;

#define NEG_SLOPE 0.01f

// ---------------------------------------------------------------------------
// Degree / normalization
// ---------------------------------------------------------------------------
__global__ void deg_init_kernel(float* __restrict__ deg, int n) {
    int i = blockIdx.x * blockDim.x + threadIdx.x;
    if (i < n) deg[i] = 1.0f;               // self-loop contributes 1
}

__global__ void deg_accum_kernel(const int* __restrict__ dst, float* __restrict__ deg, int e) {
    int i = blockIdx.x * blockDim.x + threadIdx.x;
    if (i < e) atomicAdd(&deg[dst[i]], 1.0f);
}

__global__ void dinv_kernel(float* __restrict__ deg, int n) {
    int i = blockIdx.x * blockDim.x + threadIdx.x;
    if (i < n) deg[i] = rsqrtf(deg[i]);     // deg >= 1 always (self-loops)
}

// ---------------------------------------------------------------------------
// WMMA f32 GEMM:  out[i][0:64] = (A[i][0:K] @ W[K][64]) * dinv[i]
// One wave per 16-row tile; 4 accumulators cover all 64 output columns.
// Uses V_WMMA_F32_16X16X4_F32 (wave32, EXEC all-ones within active waves).
// A-frag layout (16x4 f32): lane l -> row l%16, K pair at k + 2*(l/16).
// B-frag layout (4x16 f32): lane l -> col l%16, K pair at k + 2*(l/16).
// D layout (16x16 f32):     vgpr r, lane l -> row r + 8*(l/16), col l%16.
// ---------------------------------------------------------------------------
template <int K>
__global__ void gemm_scale_kernel(const float* __restrict__ A,
                                  const float* __restrict__ W,   // [K,64]
                                  const float* __restrict__ dinv,
                                  float* __restrict__ out,       // [N,64]
                                  int nRowTiles) {
    int wave = (int)((blockIdx.x * blockDim.x + threadIdx.x) >> 5);
    if (wave >= nRowTiles) return;          // wave-uniform exit, EXEC stays full
    int lane = threadIdx.x & 31;
    int m    = lane & 15;                   // row within tile (A) / col (B,D)
    int kh   = lane >> 4;                   // 0 or 1 -> K offset 2*kh / row offset 8*kh

    v8f acc0 = {}, acc1 = {}, acc2 = {}, acc3 = {};

    const float* arow = A + (size_t)(wave * 16 + m) * K + 2 * kh;

    for (int k = 0; k < K; k += 4) {
        v2f a;
        a.x = arow[k + 0];
        a.y = arow[k + 1];

        const float* w0 = W + (size_t)(k + 2 * kh) * 64 + m;
        const float* w1 = w0 + 64;
        v2f b0, b1, b2, b3;
        b0.x = w0[0];   b0.y = w1[0];
        b1.x = w0[16];  b1.y = w1[16];
        b2.x = w0[32];  b2.y = w1[32];
        b3.x = w0[48];  b3.y = w1[48];

        acc0 = __builtin_amdgcn_wmma_f32_16x16x4_f32(false, a, false, b0, (short)0, acc0, false, false);
        acc1 = __builtin_amdgcn_wmma_f32_16x16x4_f32(false, a, false, b1, (short)0, acc1, false, false);
        acc2 = __builtin_amdgcn_wmma_f32_16x16x4_f32(false, a, false, b2, (short)0, acc2, false, false);
        acc3 = __builtin_amdgcn_wmma_f32_16x16x4_f32(false, a, false, b3, (short)0, acc3, false, false);
    }

    int rowBase = wave * 16 + 8 * kh;
#pragma unroll
    for (int r = 0; r < 8; ++r) {
        int row = rowBase + r;
        float s = dinv[row];
        float* orow = out + (size_t)row * 64 + m;
        orow[0]  = acc0[r] * s;
        orow[16] = acc1[r] * s;
        orow[32] = acc2[r] * s;
        orow[48] = acc3[r] * s;
    }
}

// ---------------------------------------------------------------------------
// acc := hs  (self-loop initialization), float4 vectorized
// ---------------------------------------------------------------------------
__global__ void copy_init_kernel(const float4* __restrict__ src, float4* __restrict__ dst, int n4) {
    int i = blockIdx.x * blockDim.x + threadIdx.x;
    if (i < n4) dst[i] = src[i];
}

// ---------------------------------------------------------------------------
// Edge scatter: acc[dst] += hs[src].  16 threads/edge, float4 gather,
// 4 global_atomic_add_f32 each.  Feature buffers live in L2 (25.6MB << 192MB).
// ---------------------------------------------------------------------------
__global__ void edge_scatter_kernel(const int* __restrict__ src, const int* __restrict__ dst,
                                    const float* __restrict__ hs, float* __restrict__ acc, int e) {
    int t = blockIdx.x * blockDim.x + threadIdx.x;
    int ei = t >> 4;
    if (ei >= e) return;
    int c = (t & 15) * 4;
    int s = src[ei];
    int d = dst[ei];
    float4 v = *(const float4*)(hs + (size_t)s * 64 + c);
    float* p = acc + (size_t)d * 64 + c;
    atomicAdd(p + 0, v.x);
    atomicAdd(p + 1, v.y);
    atomicAdd(p + 2, v.z);
    atomicAdd(p + 3, v.w);
}

// ---------------------------------------------------------------------------
// Finalize: out = leaky_relu(acc * dinv[row] + b[col])
// ---------------------------------------------------------------------------
__global__ void finalize_kernel(const float* __restrict__ acc, const float* __restrict__ dinv,
                                const float* __restrict__ b, float* __restrict__ out, int total) {
    int i = blockIdx.x * blockDim.x + threadIdx.x;
    if (i >= total) return;
    int row = i >> 6;
    int col = i & 63;
    float v = fmaf(acc[i], dinv[row], b[col]);
    out[i] = (v > 0.0f) ? v : NEG_SLOPE * v;
}

// ---------------------------------------------------------------------------
// Select + MLP: h_sel = h[idx[j]], y = sigmoid(h_sel @ Wm + bm)
// One thread per selected row: 64 loads, 5x64 FMAs -- negligible work.
// ---------------------------------------------------------------------------
__global__ void select_mlp_kernel(const float* __restrict__ h, const int* __restrict__ idx,
                                  const float* __restrict__ Wm,  // [64,5]
                                  const float* __restrict__ bm,  // [5]
                                  float* __restrict__ out_hsel,  // [M,64]
                                  float* __restrict__ out_y,     // [M,5]
                                  int m) {
    int j = blockIdx.x * blockDim.x + threadIdx.x;
    if (j >= m) return;
    int row = idx[j];
    const float* hr = h + (size_t)row * 64;
    float* o = out_hsel + (size_t)j * 64;
    float acc[5];
#pragma unroll
    for (int c = 0; c < 5; ++c) acc[c] = bm[c];
    for (int k = 0; k < 64; ++k) {
        float v = hr[k];
        o[k] = v;
#pragma unroll
        for (int c = 0; c < 5; ++c) acc[c] = fmaf(v, Wm[k * 5 + c], acc[c]);
    }
    float* oy = out_y + (size_t)j * 5;
#pragma unroll
    for (int c = 0; c < 5; ++c) oy[c] = 1.0f / (1.0f + __expf(-acc[c]));
}

// ---------------------------------------------------------------------------
// Host-side launch
// ---------------------------------------------------------------------------
extern "C" void kernel_launch(void* const* d_in, const int* in_sizes, int n_in,
                              void* d_out, int out_size, void* d_ws, size_t ws_size,
                              hipStream_t stream) {
    const float* x    = (const float*)d_in[0];   // [N,128]
    const int*   eidx = (const int*)d_in[1];     // [2,E]
    const int*   idx  = (const int*)d_in[2];     // [M]
    const float* W0   = (const float*)d_in[3];   // [128,64]
    const float* b0   = (const float*)d_in[4];   // [64]
    const float* W1   = (const float*)d_in[5];   // [64,64]
    const float* b1   = (const float*)d_in[6];   // [64]
    const float* Wm   = (const float*)d_in[7];   // [64,5]
    const float* bm   = (const float*)d_in[8];   // [5]

    const int N = in_sizes[0] / 128;
    const int E = in_sizes[1] / 2;
    const int M = in_sizes[2];

    const int* src = eidx;
    const int* dst = eidx + E;

    // Workspace layout: dinv[N] | ws1[N*64] | ws2[N*64]
    float* dinv = (float*)d_ws;
    size_t nAl  = ((size_t)N + 255) & ~(size_t)255;
    float* ws1  = dinv + nAl;
    float* ws2  = ws1 + (size_t)N * 64;

    float* out_hsel = (float*)d_out;
    float* out_y    = out_hsel + (size_t)M * 64;

    const int BT = 256;
    int nBlkN   = (N + BT - 1) / BT;
    int nBlkE   = (E + BT - 1) / BT;
    int rowTiles = (N + 15) / 16;
    int nBlkGemm = (rowTiles * 32 + BT - 1) / BT;
    int feat4    = N * 16;                       // N*64 floats / 4
    int nBlkC    = (feat4 + BT - 1) / BT;
    int nBlkScat = (E * 16 + BT - 1) / BT;
    int featTot  = N * 64;
    int nBlkF    = (featTot + BT - 1) / BT;
    int nBlkM    = (M + BT - 1) / BT;

    // 1) symmetric normalization coefficients
    deg_init_kernel<<<nBlkN, BT, 0, stream>>>(dinv, N);
    deg_accum_kernel<<<nBlkE, BT, 0, stream>>>(dst, dinv, E);
    dinv_kernel<<<nBlkN, BT, 0, stream>>>(dinv, N);

    // 2) layer 1:  hs = (x@W0)*dinv  -> acc = hs + scatter -> act = lrelu(acc*dinv+b0)
    gemm_scale_kernel<128><<<nBlkGemm, BT, 0, stream>>>(x, W0, dinv, ws1, rowTiles);
    copy_init_kernel<<<nBlkC, BT, 0, stream>>>((const float4*)ws1, (float4*)ws2, feat4);
    edge_scatter_kernel<<<nBlkScat, BT, 0, stream>>>(src, dst, ws1, ws2, E);
    finalize_kernel<<<nBlkF, BT, 0, stream>>>(ws2, dinv, b0, ws1, featTot);

    // 3) layer 2:  hs2 = (act@W1)*dinv -> acc2 -> hfinal
    gemm_scale_kernel<64><<<nBlkGemm, BT, 0, stream>>>(ws1, W1, dinv, ws2, rowTiles);
    copy_init_kernel<<<nBlkC, BT, 0, stream>>>((const float4*)ws2, (float4*)ws1, feat4);
    edge_scatter_kernel<<<nBlkScat, BT, 0, stream>>>(src, dst, ws2, ws1, E);
    finalize_kernel<<<nBlkF, BT, 0, stream>>>(ws1, dinv, b1, ws2, featTot);

    // 4) select + MLP head
    select_mlp_kernel<<<nBlkM, BT, 0, stream>>>(ws2, idx, Wm, bm, out_hsel, out_y, M);
}